// PIVAE_40973988004430
// MI455X (gfx1250) — compile-verified
//
#include <hip/hip_runtime.h>
#include <hip/hip_bf16.h>
#include <math.h>

// ---------------- problem dims ----------------
#define NB   2048   // batch
#define NE   256    // evals
#define NC   100    // centers
#define ND1  20
#define ND2  20
#define NO   100    // out_dims
#define NH1  128
#define NH2  64
#define NZ   20

typedef __attribute__((ext_vector_type(2))) float v2f;
typedef __attribute__((ext_vector_type(8))) float v8f;

// D = A(16x4) * B(4x16) + C   (f32 WMMA, wave32)
#define WMMA4(a, b, c) \
  __builtin_amdgcn_wmma_f32_16x16x4_f32(false, (a), false, (b), (short)0, (c), false, false)

// =====================================================================
// Kernel 1: VAE over beta_W rows.  One 16-row tile per wave.
//   he1 = relu(bw @ eW1 + eb1)        K=100 N=128
//   he2 = relu(he1 @ eW2 + eb2)       K=128 N=64
//   mu/lv = he2 @ Wmu/Wlv + b         K=64  N=20 (pad 32)
//   z  = mu + exp(0.5 lv)*eps
//   hd1 = relu(z @ dW1 + db1)         K=20  N=64
//   hd2 = relu(hd1 @ dW2 + db2)       K=64  N=128
//   recon = hd2 @ dW3 + db3           K=128 N=100 (pad 112)
// =====================================================================
__global__ __launch_bounds__(64) void pivae_vae_kernel(
    const float* __restrict__ betaW, const float* __restrict__ eps,
    const float* __restrict__ eW1, const float* __restrict__ eb1,
    const float* __restrict__ eW2, const float* __restrict__ eb2,
    const float* __restrict__ Wmu, const float* __restrict__ bmu,
    const float* __restrict__ Wlv, const float* __restrict__ blv,
    const float* __restrict__ dW1, const float* __restrict__ db1,
    const float* __restrict__ dW2, const float* __restrict__ db2,
    const float* __restrict__ dW3, const float* __restrict__ db3,
    float* __restrict__ muOut, float* __restrict__ lvOut,
    float* __restrict__ recon)
{
  __shared__ float buf[2][2][16][128];   // [wave][pingpong][m][n]  32 KB

  const int wave = threadIdx.x >> 5;
  const int lane = threadIdx.x & 31;
  const int hi   = lane >> 4;            // half-wave: 0|1
  const int ln   = lane & 15;            // n (B/C/D) or m (A) within half
  const int kofs = hi << 1;              // f32 A/B: K offset per half-wave
  const int m0   = (blockIdx.x * 2 + wave) * 16;

  float (*Ab)[128] = buf[wave][0];
  float (*Bb)[128] = buf[wave][1];

  // load beta_W tile, zero-pad cols 100..127
  for (int i = lane; i < 16 * 128; i += 32) {
    int m = i >> 7, n = i & 127;
    Ab[m][n] = (n < NO) ? betaW[(m0 + m) * NO + n] : 0.0f;
  }

  // ---- enc1 ----
  for (int nt = 0; nt < 8; ++nt) {
    int n = nt * 16 + ln;
    v8f acc; float bias = eb1[n];
    #pragma unroll
    for (int v = 0; v < 8; ++v) acc[v] = bias;
    for (int k0 = 0; k0 < NC; k0 += 4) {
      int kk = k0 + kofs;
      v2f a, bfr;
      a.x = Ab[ln][kk];          a.y = Ab[ln][kk + 1];
      bfr.x = eW1[kk * NH1 + n]; bfr.y = eW1[(kk + 1) * NH1 + n];
      acc = WMMA4(a, bfr, acc);
    }
    #pragma unroll
    for (int v = 0; v < 8; ++v) Bb[v + 8 * hi][n] = fmaxf(acc[v], 0.0f);
  }

  // ---- enc2 ----
  for (int nt = 0; nt < 4; ++nt) {
    int n = nt * 16 + ln;
    v8f acc; float bias = eb2[n];
    #pragma unroll
    for (int v = 0; v < 8; ++v) acc[v] = bias;
    for (int k0 = 0; k0 < NH1; k0 += 4) {
      int kk = k0 + kofs;
      v2f a, bfr;
      a.x = Bb[ln][kk];          a.y = Bb[ln][kk + 1];
      bfr.x = eW2[kk * NH2 + n]; bfr.y = eW2[(kk + 1) * NH2 + n];
      acc = WMMA4(a, bfr, acc);
    }
    #pragma unroll
    for (int v = 0; v < 8; ++v) Ab[v + 8 * hi][n] = fmaxf(acc[v], 0.0f);
  }

  // ---- mu / logvar / z ----
  for (int nt = 0; nt < 2; ++nt) {
    int n = nt * 16 + ln;
    bool valid = (n < NZ);
    v8f am, al;
    float bm = valid ? bmu[n] : 0.0f;
    float bl = valid ? blv[n] : 0.0f;
    #pragma unroll
    for (int v = 0; v < 8; ++v) { am[v] = bm; al[v] = bl; }
    for (int k0 = 0; k0 < NH2; k0 += 4) {
      int kk = k0 + kofs;
      v2f a, bm2, bl2;
      a.x = Ab[ln][kk]; a.y = Ab[ln][kk + 1];
      bm2.x = valid ? Wmu[kk * NZ + n] : 0.0f;
      bm2.y = valid ? Wmu[(kk + 1) * NZ + n] : 0.0f;
      bl2.x = valid ? Wlv[kk * NZ + n] : 0.0f;
      bl2.y = valid ? Wlv[(kk + 1) * NZ + n] : 0.0f;
      am = WMMA4(a, bm2, am);
      al = WMMA4(a, bl2, al);
    }
    #pragma unroll
    for (int v = 0; v < 8; ++v) {
      int row = v + 8 * hi;
      float z = 0.0f;
      if (valid) {
        int gi = (m0 + row) * NZ + n;
        muOut[gi] = am[v];
        lvOut[gi] = al[v];
        z = am[v] + expf(0.5f * al[v]) * eps[gi];
      }
      Bb[row][n] = z;                       // cols 20..31 zero-padded
    }
  }

  // ---- dec1 ----
  for (int nt = 0; nt < 4; ++nt) {
    int n = nt * 16 + ln;
    v8f acc; float bias = db1[n];
    #pragma unroll
    for (int v = 0; v < 8; ++v) acc[v] = bias;
    for (int k0 = 0; k0 < NZ; k0 += 4) {
      int kk = k0 + kofs;
      v2f a, bfr;
      a.x = Bb[ln][kk];          a.y = Bb[ln][kk + 1];
      bfr.x = dW1[kk * NH2 + n]; bfr.y = dW1[(kk + 1) * NH2 + n];
      acc = WMMA4(a, bfr, acc);
    }
    #pragma unroll
    for (int v = 0; v < 8; ++v) Ab[v + 8 * hi][n] = fmaxf(acc[v], 0.0f);
  }

  // ---- dec2 ----
  for (int nt = 0; nt < 8; ++nt) {
    int n = nt * 16 + ln;
    v8f acc; float bias = db2[n];
    #pragma unroll
    for (int v = 0; v < 8; ++v) acc[v] = bias;
    for (int k0 = 0; k0 < NH2; k0 += 4) {
      int kk = k0 + kofs;
      v2f a, bfr;
      a.x = Ab[ln][kk];          a.y = Ab[ln][kk + 1];
      bfr.x = dW2[kk * NH1 + n]; bfr.y = dW2[(kk + 1) * NH1 + n];
      acc = WMMA4(a, bfr, acc);
    }
    #pragma unroll
    for (int v = 0; v < 8; ++v) Bb[v + 8 * hi][n] = fmaxf(acc[v], 0.0f);
  }

  // ---- dec3 -> recon ----
  for (int nt = 0; nt < 7; ++nt) {
    int n = nt * 16 + ln;
    bool valid = (n < NO);
    v8f acc; float bias = valid ? db3[n] : 0.0f;
    #pragma unroll
    for (int v = 0; v < 8; ++v) acc[v] = bias;
    for (int k0 = 0; k0 < NH1; k0 += 4) {
      int kk = k0 + kofs;
      v2f a, bfr;
      a.x = Bb[ln][kk]; a.y = Bb[ln][kk + 1];
      bfr.x = valid ? dW3[kk * NO + n] : 0.0f;
      bfr.y = valid ? dW3[(kk + 1) * NO + n] : 0.0f;
      acc = WMMA4(a, bfr, acc);
    }
    #pragma unroll
    for (int v = 0; v < 8; ++v)
      if (valid) recon[(m0 + v + 8 * hi) * NO + n] = acc[v];
  }
}

// =====================================================================
// Kernel 2: fused RBF -> tanh MLP -> y1/y2 contractions.
// 4 blocks per batch, 4 waves per block, 16 E-rows per wave.
// phi_x is never written to memory: final D-fragments are dotted against
// beta_W[b] / recon[b] in-register and reduced with wave32 shuffles.
// =====================================================================
__global__ __launch_bounds__(128) void pivae_phi_kernel(
    const float* __restrict__ x, const float* __restrict__ alphaP,
    const float* __restrict__ centers,
    const float* __restrict__ W1, const float* __restrict__ b1,
    const float* __restrict__ W2, const float* __restrict__ b2,
    const float* __restrict__ W3, const float* __restrict__ b3,
    const float* __restrict__ betaW, const float* __restrict__ betaB,
    const float* __restrict__ recon,
    float* __restrict__ y1, float* __restrict__ y2)
{
  __shared__ float W1s[NC][32];       // pad N 20->32
  __shared__ float W2s[ND1][32];
  __shared__ float W3s[ND2][112];     // pad N 100->112
  __shared__ float b1s[32], b2s[32], b3s[112];
  __shared__ float cens[NC][2];
  __shared__ float bws[112], recs[112];
  __shared__ float h1s[4][16][32];
  __shared__ float h2s[4][16][32];

  const int tid  = threadIdx.x;
  const int bat  = blockIdx.x >> 2;
  const int wave = tid >> 5;
  const int lane = tid & 31;
  const int hi   = lane >> 4;
  const int ln   = lane & 15;
  const int kofs = hi << 1;
  const int e0   = ((blockIdx.x & 3) * 4 + wave) * 16;

  for (int i = tid; i < NC * 32; i += 128)  { int k = i >> 5, n = i & 31;     W1s[k][n] = (n < ND1) ? W1[k * ND1 + n] : 0.0f; }
  for (int i = tid; i < ND1 * 32; i += 128) { int k = i >> 5, n = i & 31;     W2s[k][n] = (n < ND2) ? W2[k * ND2 + n] : 0.0f; }
  for (int i = tid; i < ND2 * 112; i += 128){ int k = i / 112, n = i % 112;   W3s[k][n] = (n < NO)  ? W3[k * NO + n]  : 0.0f; }
  if (tid < 32) {
    b1s[tid] = (tid < ND1) ? b1[tid] : 0.0f;
    b2s[tid] = (tid < ND2) ? b2[tid] : 0.0f;
  }
  for (int i = tid; i < 112; i += 128) {
    b3s[i]  = (i < NO) ? b3[i] : 0.0f;
    bws[i]  = (i < NO) ? betaW[bat * NO + i] : 0.0f;
    recs[i] = (i < NO) ? recon[bat * NO + i] : 0.0f;
  }
  for (int i = tid; i < NC * 2; i += 128) cens[i >> 1][i & 1] = centers[i];
  __syncthreads();

  const float alpha = alphaP[0];
  const float px0 = x[(bat * NE + e0 + ln) * 2 + 0];   // this lane's row m=ln
  const float px1 = x[(bat * NE + e0 + ln) * 2 + 1];

  // ---- GEMM1: feat(16x100) @ W1 -> tanh -> h1 (A computed in-register) ----
  {
    v8f acc0, acc1;
    float bi0 = b1s[ln], bi1 = b1s[16 + ln];
    #pragma unroll
    for (int v = 0; v < 8; ++v) { acc0[v] = bi0; acc1[v] = bi1; }
    for (int k0 = 0; k0 < NC; k0 += 4) {
      int kk = k0 + kofs;
      v2f a, bf0, bf1;
      float dx = px0 - cens[kk][0], dy = px1 - cens[kk][1];
      a.x = __expf(-alpha * (dx * dx + dy * dy));
      dx = px0 - cens[kk + 1][0]; dy = px1 - cens[kk + 1][1];
      a.y = __expf(-alpha * (dx * dx + dy * dy));
      bf0.x = W1s[kk][ln];      bf0.y = W1s[kk + 1][ln];
      bf1.x = W1s[kk][16 + ln]; bf1.y = W1s[kk + 1][16 + ln];
      acc0 = WMMA4(a, bf0, acc0);
      acc1 = WMMA4(a, bf1, acc1);
    }
    #pragma unroll
    for (int v = 0; v < 8; ++v) {
      h1s[wave][v + 8 * hi][ln]      = tanhf(acc0[v]);
      h1s[wave][v + 8 * hi][16 + ln] = tanhf(acc1[v]);
    }
  }

  // ---- GEMM2: h1(16x20) @ W2 -> tanh -> h2 ----
  {
    v8f acc0, acc1;
    float bi0 = b2s[ln], bi1 = b2s[16 + ln];
    #pragma unroll
    for (int v = 0; v < 8; ++v) { acc0[v] = bi0; acc1[v] = bi1; }
    for (int k0 = 0; k0 < ND1; k0 += 4) {
      int kk = k0 + kofs;
      v2f a, bf0, bf1;
      a.x = h1s[wave][ln][kk]; a.y = h1s[wave][ln][kk + 1];
      bf0.x = W2s[kk][ln];      bf0.y = W2s[kk + 1][ln];
      bf1.x = W2s[kk][16 + ln]; bf1.y = W2s[kk + 1][16 + ln];
      acc0 = WMMA4(a, bf0, acc0);
      acc1 = WMMA4(a, bf1, acc1);
    }
    #pragma unroll
    for (int v = 0; v < 8; ++v) {
      h2s[wave][v + 8 * hi][ln]      = tanhf(acc0[v]);
      h2s[wave][v + 8 * hi][16 + ln] = tanhf(acc1[v]);
    }
  }

  // ---- GEMM3: h2(16x20) @ W3 -> phi fragments -> dual contraction ----
  float s1[8], s2[8];
  #pragma unroll
  for (int v = 0; v < 8; ++v) { s1[v] = 0.0f; s2[v] = 0.0f; }

  for (int nt = 0; nt < 7; ++nt) {
    int n = nt * 16 + ln;
    v8f acc; float bias = b3s[n];
    #pragma unroll
    for (int v = 0; v < 8; ++v) acc[v] = bias;
    for (int k0 = 0; k0 < ND2; k0 += 4) {
      int kk = k0 + kofs;
      v2f a, bfr;
      a.x = h2s[wave][ln][kk]; a.y = h2s[wave][ln][kk + 1];
      bfr.x = W3s[kk][n];      bfr.y = W3s[kk + 1][n];
      acc = WMMA4(a, bfr, acc);
    }
    float bw = bws[n], rc = recs[n];
    #pragma unroll
    for (int v = 0; v < 8; ++v) { s1[v] += acc[v] * bw; s2[v] += acc[v] * rc; }
  }

  // reduce across the 16 lanes of each half-wave (rows v+8*hi)
  const float bbias = betaB[bat];
  #pragma unroll
  for (int v = 0; v < 8; ++v) {
    float a1 = s1[v], a2 = s2[v];
    for (int m = 1; m < 16; m <<= 1) {
      a1 += __shfl_xor(a1, m, 32);
      a2 += __shfl_xor(a2, m, 32);
    }
    if (ln == 0) {
      int row = e0 + v + 8 * hi;
      y1[bat * NE + row] = a1 + bbias;
      y2[bat * NE + row] = a2 + bbias;
    }
  }
}

// =====================================================================
extern "C" void kernel_launch(void* const* d_in, const int* in_sizes, int n_in,
                              void* d_out, int out_size, void* d_ws, size_t ws_size,
                              hipStream_t stream) {
  const float* x       = (const float*)d_in[0];
  const float* eps     = (const float*)d_in[1];
  const float* alpha   = (const float*)d_in[2];
  const float* centers = (const float*)d_in[3];
  const float* W1  = (const float*)d_in[4];   const float* b1  = (const float*)d_in[5];
  const float* W2  = (const float*)d_in[6];   const float* b2  = (const float*)d_in[7];
  const float* W3  = (const float*)d_in[8];   const float* b3  = (const float*)d_in[9];
  const float* bW  = (const float*)d_in[10];  const float* bB  = (const float*)d_in[11];
  const float* eW1 = (const float*)d_in[12];  const float* eb1 = (const float*)d_in[13];
  const float* eW2 = (const float*)d_in[14];  const float* eb2 = (const float*)d_in[15];
  const float* Wmu = (const float*)d_in[16];  const float* bmu = (const float*)d_in[17];
  const float* Wlv = (const float*)d_in[18];  const float* blv = (const float*)d_in[19];
  const float* dW1 = (const float*)d_in[20];  const float* db1 = (const float*)d_in[21];
  const float* dW2 = (const float*)d_in[22];  const float* db2 = (const float*)d_in[23];
  const float* dW3 = (const float*)d_in[24];  const float* db3 = (const float*)d_in[25];

  float* out  = (float*)d_out;
  float* y1   = out;                       // (B,E)
  float* y2   = out + (size_t)NB * NE;     // (B,E)
  float* muO  = out + (size_t)2 * NB * NE; // (B,Z)
  float* lvO  = muO + (size_t)NB * NZ;     // (B,Z)
  float* recon = (float*)d_ws;             // (B,O) scratch

  // VAE first (produces recon), then fused PHI + contractions.
  pivae_vae_kernel<<<64, 64, 0, stream>>>(
      bW, eps, eW1, eb1, eW2, eb2, Wmu, bmu, Wlv, blv,
      dW1, db1, dW2, db2, dW3, db3, muO, lvO, recon);

  pivae_phi_kernel<<<NB * 4, 128, 0, stream>>>(
      x, alpha, centers, W1, b1, W2, b2, W3, b3,
      bW, bB, recon, y1, y2);
}